// Module_11931419149205
// MI455X (gfx1250) — compile-verified
//
#include <hip/hip_runtime.h>

// Problem dims (match reference)
#define B 8
#define N 512
#define T 48
#define O 10
#define NN (N * N)
#define BN (B * N)
#define EB 8              // e-chunks for the synapse kernel
#define ECH (N / EB)      // 64 presynaptic neurons per chunk

// Time constants
#define ALPHA_R   0.9900498f
#define ALPHA_X   0.9355070f
#define ALPHA_P   0.9512294f
#define ALPHA_D   0.9672161f
#define ALPHA_OUT 0.9512294f
#define T_REF     5.0f
#define WMAX_TOT  2.0f

typedef __attribute__((ext_vector_type(2))) float v2f;
typedef __attribute__((ext_vector_type(8))) float v8f;

// ---------------------------------------------------------------------------
// Init: fused signed weights + 1-byte delay index (replaces 3x[N,N] dmap)
// ---------------------------------------------------------------------------
__global__ void k_init_wd(const float* __restrict__ w,
                          const float* __restrict__ w_signs,
                          const float* __restrict__ dmap,
                          float* __restrict__ W,
                          unsigned char* __restrict__ didx) {
  int idx = blockIdx.x * blockDim.x + threadIdx.x;
  if (idx >= NN) return;
  int e = idx / N;
  W[idx] = w_signs[e] * fabsf(w[idx]);
  unsigned char d = 3;                       // 3 == "no connection" slot
  if (dmap[idx] > 0.f) d = 0;
  else if (dmap[NN + idx] > 0.f) d = 1;
  else if (dmap[2 * NN + idx] > 0.f) d = 2;
  didx[idx] = d;
}

__global__ void k_init_state(float* __restrict__ w_stdp, const float* __restrict__ mem0,
                             float* __restrict__ mem, float* __restrict__ w_p,
                             float* __restrict__ x_bar, float* __restrict__ u_pot,
                             float* __restrict__ u_dep, float* __restrict__ refr) {
  size_t total = (size_t)B * NN;
  for (size_t i = (size_t)blockIdx.x * blockDim.x + threadIdx.x; i < total;
       i += (size_t)gridDim.x * blockDim.x) {
    w_stdp[i] = 1.0f;
    if (i < BN) {
      mem[i] = mem0[i];
      w_p[i] = 0.f; x_bar[i] = 0.f; u_pot[i] = 0.f; u_dep[i] = 0.f; refr[i] = 0.f;
    }
  }
}

// Zero-pad w_out [N,O] -> [N,16] so the WMMA B-fragment loads are branch-free
__global__ void k_pad_wout(const float* __restrict__ w_out, float* __restrict__ w_out_pad) {
  int idx = blockIdx.x * blockDim.x + threadIdx.x;   // over N*16
  if (idx >= N * 16) return;
  int k = idx >> 4, c = idx & 15;
  w_out_pad[idx] = (c < O) ? w_out[(size_t)k * O + c] : 0.f;
}

// ---------------------------------------------------------------------------
// Step kernel A: mem += input, threshold spike, record out history
// ---------------------------------------------------------------------------
__global__ void k_step_a(const float* __restrict__ mem, const float* __restrict__ inputs,
                         float* __restrict__ mem_tmp, float* __restrict__ out_hist, int t) {
  int idx = blockIdx.x * blockDim.x + threadIdx.x;
  if (idx >= BN) return;
  int b = idx / N, n = idx % N;
  float mv = mem[idx] + inputs[((size_t)b * T + t) * N + n];
  mem_tmp[idx] = mv;
  out_hist[(size_t)t * BN + idx] = (mv - 1.0f >= 0.0f) ? 1.0f : 0.0f;
}

// ---------------------------------------------------------------------------
// Step kernel B: the [B,N,N] synapse/STDP pass.
// One block = (batch b, e-chunk). 128 threads x float4 cover all 512 post 'o'.
// Per-presynaptic scalars (spike, 1+w_p, x_bar at each delay) broadcast via LDS.
// All [N,N] operands are L2-resident (192 MB L2; total state ~12 MB).
// ---------------------------------------------------------------------------
__global__ void __launch_bounds__(128)
k_step_syn(const float* __restrict__ W, const unsigned char* __restrict__ didx,
           const float* __restrict__ A_p, const float* __restrict__ A_d,
           const float* __restrict__ out_hist, const float* __restrict__ wp_hist,
           const float* __restrict__ xb_hist, const float* __restrict__ u_pot,
           const float* __restrict__ u_dep, float* __restrict__ w_stdp,
           float* __restrict__ syn_part, int t) {
  const int b   = blockIdx.x / EB;
  const int eb  = blockIdx.x % EB;
  const int e0  = eb * ECH;
  const int tid = threadIdx.x;     // 0..127
  const int o0  = tid * 4;

  __shared__ float sh_a [4][ECH];  // out*(1+w_p) per delay; slot 3 == 0 (no conn)
  __shared__ float sh_ob[4][ECH];  // out per delay
  __shared__ float sh_xb[4][ECH];  // x_bar per delay

  if (tid < ECH) {
    const int dly[3] = {1, 2, 4};
    int e = e0 + tid;
#pragma unroll
    for (int i = 0; i < 3; i++) {
      int d = dly[i];
      float ob = (t - d     >= 0) ? out_hist[(size_t)(t - d)     * BN + b * N + e] : 0.f;
      float wp = (t - 1 - d >= 0) ? wp_hist [(size_t)(t - 1 - d) * BN + b * N + e] : 0.f;
      float xb = (t - 1 - d >= 0) ? xb_hist [(size_t)(t - 1 - d) * BN + b * N + e] : 0.f;
      sh_ob[i][tid] = ob;
      sh_a [i][tid] = ob * (1.0f + wp);
      sh_xb[i][tid] = xb;
    }
    sh_ob[3][tid] = 0.f; sh_a[3][tid] = 0.f; sh_xb[3][tid] = 0.f;
  }
  __syncthreads();

  // Per-postsynaptic scalars for this thread's 4 columns
  float potf[4], depf[4], acc[4] = {0.f, 0.f, 0.f, 0.f};
#pragma unroll
  for (int j = 0; j < 4; j++) {
    int o = o0 + j;
    float out_t = out_hist[(size_t)t * BN + b * N + o];
    potf[j] = out_t * fmaxf(u_pot[b * N + o], 0.f);   // out * relu(u_pot)
    depf[j] = fmaxf(u_dep[b * N + o], 0.f);           // relu(u_dep)
  }

  for (int ee = 0; ee < ECH; ++ee) {
    const int e = e0 + ee;
    const float4 Wv        = *(const float4*)(W    + (size_t)e * N + o0);
    const unsigned int dv  = *(const unsigned int*)(didx + (size_t)e * N + o0);
    const float4 Apv       = *(const float4*)(A_p  + (size_t)e * N + o0);
    const float4 Adv       = *(const float4*)(A_d  + (size_t)e * N + o0);
    float* wsp             = w_stdp + ((size_t)b * N + e) * N + o0;
    float4 wsv             = *(float4*)wsp;
    if (ee + 1 < ECH)  // gfx1250 global_prefetch_b8 for the next w_stdp row
      __builtin_prefetch(w_stdp + ((size_t)b * N + e + 1) * N + o0, 1, 1);

    float wj [4] = {Wv.x,  Wv.y,  Wv.z,  Wv.w};
    float apj[4] = {Apv.x, Apv.y, Apv.z, Apv.w};
    float adj[4] = {Adv.x, Adv.y, Adv.z, Adv.w};
    float wsj[4] = {wsv.x, wsv.y, wsv.z, wsv.w};
#pragma unroll
    for (int j = 0; j < 4; j++) {
      int   d      = (dv >> (8 * j)) & 0xFF;
      float wv     = wj[j];
      float ws_old = wsj[j];
      acc[j] = fmaf(sh_a[d][ee] * wv, ws_old, acc[j]);           // syn contribution
      float dpot = sh_xb[d][ee] * apj[j] * potf[j];
      float ddep = sh_ob[d][ee] * adj[j] * depf[j];
      float wmax = (wv == 0.f) ? 0.f : WMAX_TOT / fabsf(wv);
      float wn   = fminf(ws_old + dpot - ddep, wmax);
      wsj[j]     = fmaxf(wn, 0.f);
    }
    *(float4*)wsp = make_float4(wsj[0], wsj[1], wsj[2], wsj[3]);
  }

  float* sp = syn_part + ((size_t)eb * B + b) * N + o0;
  sp[0] = acc[0]; sp[1] = acc[1]; sp[2] = acc[2]; sp[3] = acc[3];
}

// ---------------------------------------------------------------------------
// Step kernel C: reduce syn partials, STP/STDP traces, membrane + refractory
// ---------------------------------------------------------------------------
__global__ void k_step_c(float* __restrict__ mem, const float* __restrict__ mem_tmp,
                         const float* __restrict__ out_hist, const float* __restrict__ syn_part,
                         float* __restrict__ w_p, float* __restrict__ x_bar,
                         float* __restrict__ u_pot, float* __restrict__ u_dep,
                         float* __restrict__ refr, const float* __restrict__ p,
                         const float* __restrict__ alpha_mem,
                         float* __restrict__ wp_hist, float* __restrict__ xb_hist, int t) {
  int idx = blockIdx.x * blockDim.x + threadIdx.x;
  if (idx >= BN) return;
  int n = idx % N;
  float mv  = mem_tmp[idx];
  float out = out_hist[(size_t)t * BN + idx];

  float syn = 0.f;
#pragma unroll
  for (int eb = 0; eb < EB; eb++) syn += syn_part[(size_t)eb * BN + idx];

  float pn  = p[n];
  float pd  = (pn < 0.f) ? 1.f : 0.f;
  float wp  = w_p[idx];
  float wpn = wp * ALPHA_R + out * pn * (1.f + pd * wp);
  w_p[idx] = wpn;
  wp_hist[(size_t)t * BN + idx] = wpn;

  float xb = ALPHA_X * x_bar[idx] + (1.f - ALPHA_X) * out;
  x_bar[idx] = xb;
  xb_hist[(size_t)t * BN + idx] = xb;

  u_pot[idx] = ALPHA_P * u_pot[idx] + (1.f - ALPHA_P) * mv;
  u_dep[idx] = ALPHA_D * u_dep[idx] + (1.f - ALPHA_D) * mv;

  float mm = alpha_mem[n] * mv + syn;
  float r  = refr[idx];
  if (out > 0.f) r = T_REF;
  if (r > 0.f) { mm = 0.f; r -= 1.f; }
  mem[idx]  = mm;
  refr[idx] = r;
}

// ---------------------------------------------------------------------------
// Readout GEMM via V_WMMA_F32_16X16X4_F32:
//   h2[T*B, 16] = out_hist[T*B, 512] x w_out_pad[512, 16]
// One wave per 16-row tile (24 waves). Branch-free inner loop: A fragment is a
// contiguous b64 load, B fragment two b32 loads from the zero-padded weights.
// A layout per ISA: lanes 0-15 hold K=0/1 in v0/v1, lanes 16-31 hold K=2/3;
// B mirrored; C is the standard 16x16 f32 accumulator layout.
// ---------------------------------------------------------------------------
__global__ void __launch_bounds__(32)
k_readout_wmma(const float* __restrict__ out_hist, const float* __restrict__ w_out_pad,
               float* __restrict__ h2) {
  const int lane    = threadIdx.x;          // 0..31
  const int rowbase = blockIdx.x * 16;      // rows are (t*B + b)
  const int m       = lane & 15;
  const int kk      = (lane < 16) ? 0 : 2;

  const float* arow = out_hist + (size_t)(rowbase + m) * N + kk;
  const float* bcol = w_out_pad + (size_t)kk * 16 + m;

  v8f c = {};
#pragma unroll 4
  for (int k0 = 0; k0 < N; k0 += 4) {
    v2f a, bf;
    a.x  = arow[k0];
    a.y  = arow[k0 + 1];
    bf.x = bcol[(size_t)k0 * 16];
    bf.y = bcol[(size_t)(k0 + 1) * 16];
    c = __builtin_amdgcn_wmma_f32_16x16x4_f32(false, a, false, bf, (short)0, c,
                                              false, false);
  }
#pragma unroll
  for (int j = 0; j < 8; j++) {
    int rm = j + ((lane < 16) ? 0 : 8);
    h2[(size_t)(rowbase + rm) * 16 + m] = c[j];
  }
}

// Leaky readout integrator over t, output transposed to [B,T,O]
__global__ void k_readout_scan(const float* __restrict__ h2, float* __restrict__ out) {
  int tid = threadIdx.x;
  if (tid >= B * O) return;
  int b = tid / O, o = tid % O;
  float acc = 0.f;
  for (int t = 0; t < T; t++) {
    acc = ALPHA_OUT * acc + h2[(size_t)(t * B + b) * 16 + o];
    out[((size_t)b * T + t) * O + o] = acc;
  }
}

// ---------------------------------------------------------------------------
extern "C" void kernel_launch(void* const* d_in, const int* in_sizes, int n_in,
                              void* d_out, int out_size, void* d_ws, size_t ws_size,
                              hipStream_t stream) {
  (void)in_sizes; (void)n_in; (void)out_size; (void)ws_size;
  const float* inputs    = (const float*)d_in[0];
  const float* w         = (const float*)d_in[1];
  const float* w_out     = (const float*)d_in[2];
  const float* p         = (const float*)d_in[3];
  const float* A_p       = (const float*)d_in[4];
  const float* A_d       = (const float*)d_in[5];
  const float* alpha_mem = (const float*)d_in[6];
  const float* mem0      = (const float*)d_in[7];
  const float* w_signs   = (const float*)d_in[8];
  const float* dmap      = (const float*)d_in[9];
  float* outp = (float*)d_out;

  char* base = (char*)d_ws;
  size_t off = 0;
  auto alloc = [&](size_t bytes) -> void* {
    void* ptr = base + off;
    off = (off + bytes + 255) & ~(size_t)255;
    return ptr;
  };
  float*         W         = (float*)alloc((size_t)NN * 4);
  unsigned char* didx      = (unsigned char*)alloc((size_t)NN);
  float*         w_stdp    = (float*)alloc((size_t)B * NN * 4);
  float*         mem       = (float*)alloc((size_t)BN * 4);
  float*         mem_tmp   = (float*)alloc((size_t)BN * 4);
  float*         w_p       = (float*)alloc((size_t)BN * 4);
  float*         x_bar     = (float*)alloc((size_t)BN * 4);
  float*         u_pot     = (float*)alloc((size_t)BN * 4);
  float*         u_dep     = (float*)alloc((size_t)BN * 4);
  float*         refr      = (float*)alloc((size_t)BN * 4);
  float*         out_hist  = (float*)alloc((size_t)T * BN * 4);
  float*         wp_hist   = (float*)alloc((size_t)T * BN * 4);
  float*         xb_hist   = (float*)alloc((size_t)T * BN * 4);
  float*         syn_part  = (float*)alloc((size_t)EB * BN * 4);
  float*         h2        = (float*)alloc((size_t)T * B * 16 * 4);
  float*         w_out_pad = (float*)alloc((size_t)N * 16 * 4);

  k_init_wd<<<(NN + 255) / 256, 256, 0, stream>>>(w, w_signs, dmap, W, didx);
  k_init_state<<<1024, 256, 0, stream>>>(w_stdp, mem0, mem, w_p, x_bar, u_pot, u_dep, refr);
  k_pad_wout<<<(N * 16 + 255) / 256, 256, 0, stream>>>(w_out, w_out_pad);

  for (int t = 0; t < T; t++) {
    k_step_a<<<(BN + 255) / 256, 256, 0, stream>>>(mem, inputs, mem_tmp, out_hist, t);
    k_step_syn<<<B * EB, 128, 0, stream>>>(W, didx, A_p, A_d, out_hist, wp_hist, xb_hist,
                                           u_pot, u_dep, w_stdp, syn_part, t);
    k_step_c<<<(BN + 255) / 256, 256, 0, stream>>>(mem, mem_tmp, out_hist, syn_part, w_p,
                                                   x_bar, u_pot, u_dep, refr, p, alpha_mem,
                                                   wp_hist, xb_hist, t);
  }

  k_readout_wmma<<<(T * B) / 16, 32, 0, stream>>>(out_hist, w_out_pad, h2);
  k_readout_scan<<<1, 128, 0, stream>>>(h2, outp);
}